// GNN_VN_Hierarchical_58385785422525
// MI455X (gfx1250) — compile-verified
//
#include <hip/hip_runtime.h>
#include <hip/hip_bf16.h>
#include <math.h>

// Problem constants (match reference)
#define NNODES 100000
#define NEDGES 1200000
#define ETOT   1300000      // NEDGES + NNODES self loops
#define H      64
#define HNUM_C 256
#define NLAYERS 2
#define NEG_SLOPE 0.2f

typedef __attribute__((ext_vector_type(2))) float v2f;
typedef __attribute__((ext_vector_type(8))) float v8f;

// ---------------------------------------------------------------------------
// float atomic max via signed/unsigned int ordering trick
__device__ __forceinline__ void atomicMaxF(float* addr, float v) {
  if (v >= 0.0f) atomicMax((int*)addr, __float_as_int(v));
  else           atomicMin((unsigned int*)addr, __float_as_uint(v));
}

// ---------------------------------------------------------------------------
// h = X @ W  for X:[N,3], W:[3,64]  (initial GATConv lift; K=3, scalar FMA)
__global__ __launch_bounds__(256) void k_gemm_in(const float* __restrict__ x,
                                                 const float* __restrict__ w,
                                                 float* __restrict__ h) {
  int idx = blockIdx.x * 256 + threadIdx.x;
  if (idx >= NNODES * H) return;
  int n = idx >> 6, j = idx & 63;
  float x0 = x[n * 3 + 0], x1 = x[n * 3 + 1], x2 = x[n * 3 + 2];
  h[idx] = x0 * w[j] + x1 * w[H + j] + x2 * w[2 * H + j];
}

// ---------------------------------------------------------------------------
// C[N,64] = A[N,64] @ B[64,64] via V_WMMA_F32_16X16X4_F32, fp32 end-to-end.
// One wave per 16x16 output tile; 8 waves/block = 2 row-tiles x 4 col-tiles.
// N = 100000 = 6250*16 exactly -> no ragged tiles. rowTile is wave-uniform so
// the guard keeps EXEC all-ones inside live waves (WMMA requirement).
__global__ __launch_bounds__(256) void k_gemm64(const float* __restrict__ A,
                                                const float* __restrict__ B,
                                                float* __restrict__ C) {
  int tid  = threadIdx.x;
  int wv   = tid >> 5;
  int lane = tid & 31;
  int rowTile = blockIdx.x * 2 + (wv >> 2);
  int colTile = wv & 3;
  int rowBase = rowTile * 16;
  if (rowBase >= NNODES) return;            // wave-uniform
  int colBase = colTile * 16;
  int r  = lane & 15;                       // row (A) / col (B,C) within tile
  int kg = lane >> 4;                       // K-pair select (lanes 0-15 vs 16-31)

  v8f acc = {};
#pragma unroll
  for (int k0 = 0; k0 < H; k0 += 4) {
    int ka = k0 + 2 * kg;
    // A 16x4 fragment: lane half selects K pair, vgpr selects K within pair
    v2f a = *(const v2f*)(A + (size_t)(rowBase + r) * H + ka);
    v2f b;
    b.x = B[(size_t)(ka + 0) * H + colBase + r];
    b.y = B[(size_t)(ka + 1) * H + colBase + r];
    acc = __builtin_amdgcn_wmma_f32_16x16x4_f32(
        /*neg_a=*/false, a, /*neg_b=*/false, b,
        /*c_mod=*/(short)0, acc, /*reuse_a=*/false, /*reuse_b=*/false);
  }
#pragma unroll
  for (int g = 0; g < 8; ++g) {
    int m = g + 8 * kg;                     // C layout: vgpr g, lane half -> M
    C[(size_t)(rowBase + m) * H + colBase + r] = acc[g];
  }
}

// ---------------------------------------------------------------------------
// Per-node attention scores: s_src[n] = h[n]·a_src, s_dst[n] = h[n]·a_dst.
// One wave32 per node, coalesced 2-float-per-lane loads + shuffle reduction.
__global__ __launch_bounds__(256) void k_scores(const float* __restrict__ h,
                                                const float* __restrict__ asrc,
                                                const float* __restrict__ adst,
                                                float* __restrict__ s_src,
                                                float* __restrict__ s_dst) {
  int n    = (blockIdx.x * 256 + threadIdx.x) >> 5;
  int lane = threadIdx.x & 31;
  if (n >= NNODES) return;
  float h0 = h[(size_t)n * H + lane];
  float h1 = h[(size_t)n * H + 32 + lane];
  float ss = h0 * asrc[lane] + h1 * asrc[32 + lane];
  float sd = h0 * adst[lane] + h1 * adst[32 + lane];
#pragma unroll
  for (int off = 16; off > 0; off >>= 1) {
    ss += __shfl_down(ss, off);
    sd += __shfl_down(sd, off);
  }
  if (lane == 0) { s_src[n] = ss; s_dst[n] = sd; }
}

__global__ __launch_bounds__(256) void k_node_init(float* __restrict__ amax,
                                                   float* __restrict__ denom) {
  int n = blockIdx.x * 256 + threadIdx.x;
  if (n >= NNODES) return;
  amax[n]  = -INFINITY;
  denom[n] = 0.0f;
}

// ---------------------------------------------------------------------------
// Edge pass 1: alpha = leaky_relu(s_src[src]+s_dst[dst]); segment max on dst
__global__ __launch_bounds__(256) void k_edge_max(const int* __restrict__ ei,
                                                  const float* __restrict__ s_src,
                                                  const float* __restrict__ s_dst,
                                                  float* __restrict__ alpha,
                                                  float* __restrict__ amax) {
  int e = blockIdx.x * 256 + threadIdx.x;
  if (e >= ETOT) return;
  int s, d;
  if (e < NEDGES) { s = ei[e]; d = ei[NEDGES + e]; }
  else            { s = d = e - NEDGES; }
  float a = s_src[s] + s_dst[d];
  a = (a > 0.0f) ? a : NEG_SLOPE * a;
  alpha[e] = a;
  atomicMaxF(&amax[d], a);
}

// Edge pass 2: ex = exp(alpha - amax[dst]); segment sum of ex on dst
__global__ __launch_bounds__(256) void k_edge_exp(const int* __restrict__ ei,
                                                  float* __restrict__ alpha,
                                                  const float* __restrict__ amax,
                                                  float* __restrict__ denom) {
  int e = blockIdx.x * 256 + threadIdx.x;
  if (e >= ETOT) return;
  int d = (e < NEDGES) ? ei[NEDGES + e] : (e - NEDGES);
  float ex = __expf(alpha[e] - amax[d]);
  alpha[e] = ex;
  atomicAdd(&denom[d], ex);
}

// Edge pass 3: out[dst] += (ex/denom[dst]) * h[src]; wave32 per edge,
// 2 features per lane -> fully coalesced gather + f32 atomic scatter.
__global__ __launch_bounds__(256) void k_edge_scatter(const int* __restrict__ ei,
                                                      const float* __restrict__ alpha,
                                                      const float* __restrict__ denom,
                                                      const float* __restrict__ h,
                                                      float* __restrict__ out) {
  int e    = (blockIdx.x * 256 + threadIdx.x) >> 5;
  int lane = threadIdx.x & 31;
  if (e >= ETOT) return;
  int s, d;
  if (e < NEDGES) { s = ei[e]; d = ei[NEDGES + e]; }
  else            { s = d = e - NEDGES; }
  float w = alpha[e] / denom[d];
  atomicAdd(&out[(size_t)d * H + lane],      w * h[(size_t)s * H + lane]);
  atomicAdd(&out[(size_t)d * H + 32 + lane], w * h[(size_t)s * H + 32 + lane]);
}

__global__ __launch_bounds__(256) void k_bias(float* __restrict__ out,
                                              const float* __restrict__ b) {
  int idx = blockIdx.x * 256 + threadIdx.x;
  if (idx >= NNODES * H) return;
  out[idx] += b[idx & 63];
}

// ---------------------------------------------------------------------------
// vn buffer: rows 0..255 = vn_direct, row 256 = vn_root
__global__ __launch_bounds__(256) void k_init_vn(float* __restrict__ vn,
                                                 const float* __restrict__ vn_emb) {
  int idx = blockIdx.x * 256 + threadIdx.x;
  if (idx >= (HNUM_C + 1) * H) return;
  vn[idx] = vn_emb[idx & 63];
}

__global__ __launch_bounds__(256) void k_add_vn(float* __restrict__ out,
                                                const float* __restrict__ vn,
                                                const int* __restrict__ h_blocks) {
  int idx = blockIdx.x * 256 + threadIdx.x;
  if (idx >= NNODES * H) return;
  int n = idx >> 6, j = idx & 63;
  out[idx] += vn[(size_t)h_blocks[n] * H + j];
}

// segment_sum(out over h_blocks) += into vn rows 0..255, staged through LDS
// (256x64 fp32 accumulator, dynamic LDS = 64 KB; ds_add_f32 then one global
//  atomic per slot per block -> 100 global atomics per slot instead of ~390 hot)
__global__ __launch_bounds__(256) void k_segsum(const float* __restrict__ out,
                                                const int* __restrict__ h_blocks,
                                                float* __restrict__ vn,
                                                int nodesPerBlock) {
  extern __shared__ float smem[];           // HNUM_C * H floats
  for (int i = threadIdx.x; i < HNUM_C * H; i += 256) smem[i] = 0.0f;
  __syncthreads();
  int start = blockIdx.x * nodesPerBlock;
  int end   = start + nodesPerBlock;
  if (end > NNODES) end = NNODES;
  int wv = threadIdx.x >> 5, lane = threadIdx.x & 31;
  for (int n = start + wv; n < end; n += 8) {
    int b = h_blocks[n];
    atomicAdd(&smem[b * H + lane],      out[(size_t)n * H + lane]);
    atomicAdd(&smem[b * H + 32 + lane], out[(size_t)n * H + 32 + lane]);
  }
  __syncthreads();
  for (int i = threadIdx.x; i < HNUM_C * H; i += 256) {
    float v = smem[i];
    if (v != 0.0f) atomicAdd(&vn[i], v);
  }
}

// vn_direct += vn_root; vn_root = colsum(vn_direct) + vn_root  (single block)
__global__ __launch_bounds__(256) void k_vn_finalize(float* __restrict__ vn) {
  __shared__ float rootOld[H];
  __shared__ float colsum[H];
  int t = threadIdx.x;
  if (t < H) rootOld[t] = vn[HNUM_C * H + t];
  __syncthreads();
  for (int i = t; i < HNUM_C * H; i += 256) vn[i] += rootOld[i & 63];
  __syncthreads();
  if (t < H) {
    float s = 0.0f;
    for (int b = 0; b < HNUM_C; ++b) s += vn[b * H + t];
    colsum[t] = s;
  }
  __syncthreads();
  if (t < H) vn[HNUM_C * H + t] = colsum[t] + rootOld[t];
}

// out[r][j] = relu(sum_k in[r][k]*W[k][j] + b[j]) over 257 rows (direct+root)
__global__ __launch_bounds__(256) void k_vn_mlp(const float* __restrict__ in,
                                                const float* __restrict__ W,
                                                const float* __restrict__ b,
                                                float* __restrict__ outv) {
  int idx = blockIdx.x * 256 + threadIdx.x;
  if (idx >= (HNUM_C + 1) * H) return;
  int r = idx >> 6, j = idx & 63;
  float acc = b[j];
#pragma unroll 8
  for (int k = 0; k < H; ++k) acc += in[r * H + k] * W[k * H + j];
  outv[idx] = acc > 0.0f ? acc : 0.0f;
}

// ---------------------------------------------------------------------------
// Host-side: enqueue softmax-attention aggregation (h -> out) for one GATConv
static void gat_edges(hipStream_t stream, const int* ei, const float* h,
                      float* out, const float* asrc, const float* adst,
                      const float* bias, float* alpha, float* s_src,
                      float* s_dst, float* amax, float* denom) {
  k_scores<<<(NNODES + 7) / 8, 256, 0, stream>>>(h, asrc, adst, s_src, s_dst);
  k_node_init<<<(NNODES + 255) / 256, 256, 0, stream>>>(amax, denom);
  hipMemsetAsync(out, 0, (size_t)NNODES * H * sizeof(float), stream);
  k_edge_max<<<(ETOT + 255) / 256, 256, 0, stream>>>(ei, s_src, s_dst, alpha, amax);
  k_edge_exp<<<(ETOT + 255) / 256, 256, 0, stream>>>(ei, alpha, amax, denom);
  k_edge_scatter<<<(ETOT + 7) / 8, 256, 0, stream>>>(ei, alpha, denom, h, out);
  k_bias<<<(NNODES * H + 255) / 256, 256, 0, stream>>>(out, bias);
}

extern "C" void kernel_launch(void* const* d_in, const int* in_sizes, int n_in,
                              void* d_out, int out_size, void* d_ws, size_t ws_size,
                              hipStream_t stream) {
  (void)in_sizes; (void)n_in; (void)out_size; (void)ws_size;
  const float* x        = (const float*)d_in[0];
  const int*   ei       = (const int*)  d_in[1];
  const int*   h_blocks = (const int*)  d_in[2];
  /* d_in[3] h_levels unused by reference; d_in[4] h_num == 256 */
  const float* w_in     = (const float*)d_in[5];
  const float* a_src_in = (const float*)d_in[6];
  const float* a_dst_in = (const float*)d_in[7];
  const float* b_in     = (const float*)d_in[8];
  const float* w_l      = (const float*)d_in[9];
  const float* a_src_l  = (const float*)d_in[10];
  const float* a_dst_l  = (const float*)d_in[11];
  const float* b_l      = (const float*)d_in[12];
  const float* mlp_w1   = (const float*)d_in[13];
  const float* mlp_b1   = (const float*)d_in[14];
  const float* mlp_w2   = (const float*)d_in[15];
  const float* mlp_b2   = (const float*)d_in[16];
  const float* vn_emb   = (const float*)d_in[17];

  float* out   = (float*)d_out;                 // node features live here
  float* h     = (float*)d_ws;                  // [N,64]
  float* alpha = h     + (size_t)NNODES * H;    // [ETOT]
  float* s_src = alpha + ETOT;                  // [N]
  float* s_dst = s_src + NNODES;                // [N]
  float* amax  = s_dst + NNODES;                // [N]
  float* denom = amax  + NNODES;                // [N]
  float* vn    = denom + NNODES;                // [257,64] (direct rows + root)
  float* vntmp = vn    + (HNUM_C + 1) * H;      // [257,64]

  const int VN_ELEMS = (HNUM_C + 1) * H;

  // vn_direct = broadcast(vn_emb), vn_root = vn_emb
  k_init_vn<<<(VN_ELEMS + 255) / 256, 256, 0, stream>>>(vn, vn_emb);

  // initial GATConv(3 -> 64)
  k_gemm_in<<<(NNODES * H + 255) / 256, 256, 0, stream>>>(x, w_in, h);
  gat_edges(stream, ei, h, out, a_src_in, a_dst_in, b_in,
            alpha, s_src, s_dst, amax, denom);

  for (int l = 0; l < NLAYERS; ++l) {
    // out += vn_direct[h_blocks]
    k_add_vn<<<(NNODES * H + 255) / 256, 256, 0, stream>>>(out, vn, h_blocks);
    // h = out @ W_l  (fp32 WMMA)
    k_gemm64<<<(NNODES / 16 + 1) / 2, 256, 0, stream>>>(out, w_l + (size_t)l * H * H, h);
    // softmax attention aggregation -> out
    gat_edges(stream, ei, h, out, a_src_l + l * H, a_dst_l + l * H, b_l + l * H,
              alpha, s_src, s_dst, amax, denom);
    // vn_direct += segment_sum(out); then += vn_root; vn_root = colsum + vn_root
    k_segsum<<<100, 256, HNUM_C * H * sizeof(float), stream>>>(out, h_blocks, vn, 1000);
    k_vn_finalize<<<1, 256, 0, stream>>>(vn);
    // both MLPs applied to vn_direct and vn_root (rows 0..256 together)
    for (int m = 0; m < NLAYERS; ++m) {
      k_vn_mlp<<<(VN_ELEMS + 255) / 256, 256, 0, stream>>>(
          vn, mlp_w1 + (size_t)m * H * H, mlp_b1 + m * H, vntmp);
      k_vn_mlp<<<(VN_ELEMS + 255) / 256, 256, 0, stream>>>(
          vntmp, mlp_w2 + (size_t)m * H * H, mlp_b2 + m * H, vn);
    }
  }
  // final node features already in d_out
}